// BurgerDissipativeImplicitLossOperator_16939351015518
// MI455X (gfx1250) — compile-verified
//
#include <hip/hip_runtime.h>

// BurgerDissipativeImplicitLoss on a random graph (2M nodes, 4M edges):
//   temporal = (u_t - u_t1)/DT
//   local    = (u_t[dst]-u_t[src])/ea             (per edge)
//   sums/cnt/sum_ext = segment sums keyed by dst
//   spatial  = cnt>0 ? sums/max(cnt,1) : 0
//   second   = (sum_ext - 2*u_t)/DX^2
//   loss     = (temporal + spatial*u_t - MU*second) * mask
//
// Memory/atomic-bound scatter op: no MxNxK structure -> WMMA inapplicable.
// CDNA5 paths: native GLOBAL_ATOMIC_ADD_F32 (no-return) via unsafeAtomicAdd,
// global_prefetch_b8 on gather lines, NT temporal hints so ~120MB of
// single-use streams bypass L2 while the hot set (x_t + interleaved
// accumulator records, ~56MB) stays resident in the 192MB L2.
// Each edge's 3 atomics hit ONE random cacheline (16B node record).

static constexpr float kInvDT  = 1.0f / 0.01f;              // 1/DELTA_T
static constexpr float kInvDX2 = 1.0f / (0.001f * 0.001f);  // 1/DELTA_X^2
static constexpr float kMu     = 0.01f;

// Native clang vector types: required by __builtin_nontemporal_load/store
// (HIP's int2/float4 are structs and are rejected by the builtin).
typedef int   v2i __attribute__((ext_vector_type(2)));
typedef float v4f __attribute__((ext_vector_type(4)));

__global__ void zero_ws4_kernel(v4f* __restrict__ ws, int n4) {
    int i = blockIdx.x * blockDim.x + threadIdx.x;
    if (i < n4) ws[i] = (v4f)(0.f);          // global_store_b128
}

__global__ void zero_ws_tail_kernel(float* __restrict__ ws, int base, int n) {
    int i = base + blockIdx.x * blockDim.x + threadIdx.x;
    if (i < n) ws[i] = 0.f;
}

__device__ __forceinline__ void do_edge(const float* __restrict__ x_t,
                                        float* __restrict__ acc, int rec_stride,
                                        int s, int d, float ea) {
    float us = x_t[3 * (size_t)s];   // u_t[src], hot in L2
    float ud = x_t[3 * (size_t)d];   // u_t[dst]
    float local = (ud - us) / ea;
    float* rec = acc + (size_t)d * rec_stride;
    unsafeAtomicAdd(rec + 0, local);  // sums
    unsafeAtomicAdd(rec + 1, 1.0f);   // cnt
    unsafeAtomicAdd(rec + 2, us);     // sum_ext
}

// Two edges per thread: b64 index loads, one b128 edge_attr load (all NT).
__global__ void edge_scatter_kernel(const float* __restrict__ x_t,
                                    const float* __restrict__ edge_attr,
                                    const int*   __restrict__ src,
                                    const int*   __restrict__ dst,
                                    float* __restrict__ acc,
                                    int rec_stride,           // 4 (pref) or 3
                                    int n_edges) {
    int e = (blockIdx.x * blockDim.x + threadIdx.x) * 2;
    if (e >= n_edges) return;

    if (e + 1 < n_edges) {
        v2i s2 = __builtin_nontemporal_load(
            reinterpret_cast<const v2i*>(src + e));
        v2i d2 = __builtin_nontemporal_load(
            reinterpret_cast<const v2i*>(dst + e));

        // Kick the four random gather cachelines early (global_prefetch_b8).
        __builtin_prefetch(&x_t[3 * (size_t)s2.x], 0, 3);
        __builtin_prefetch(&x_t[3 * (size_t)d2.x], 0, 3);
        __builtin_prefetch(&x_t[3 * (size_t)s2.y], 0, 3);
        __builtin_prefetch(&x_t[3 * (size_t)d2.y], 0, 3);

        // edge_attr rows e, e+1 -> 16B-aligned (e even): one NT b128 load.
        v4f ea4 = __builtin_nontemporal_load(
            reinterpret_cast<const v4f*>(edge_attr + 2 * (size_t)e));

        do_edge(x_t, acc, rec_stride, s2.x, d2.x, ea4.x);
        do_edge(x_t, acc, rec_stride, s2.y, d2.y, ea4.z);
    } else {  // odd tail edge
        int s = __builtin_nontemporal_load(&src[e]);
        int d = __builtin_nontemporal_load(&dst[e]);
        float ea = __builtin_nontemporal_load(&edge_attr[2 * (size_t)e]);
        do_edge(x_t, acc, rec_stride, s, d, ea);
    }
}

__device__ __forceinline__ void node_combine(float ut, float ut1, float sm,
                                             float c, float se, float m,
                                             float* __restrict__ out, int i) {
    float temporal = (ut - ut1) * kInvDT;
    float spatial  = (c > 0.0f) ? (sm / fmaxf(c, 1.0f)) : 0.0f;
    float second   = (se - 2.0f * ut) * kInvDX2;
    float loss     = (temporal + spatial * ut - kMu * second) * m;
    __builtin_nontemporal_store(loss, &out[i]);
}

// rec_stride == 4: one NT b128 load of the whole node record.
__global__ void node_loss4_kernel(const float* __restrict__ x_t,
                                  const float* __restrict__ x_t1,
                                  const float* __restrict__ mask,
                                  const v4f* __restrict__ acc4,
                                  float* __restrict__ out, int n_nodes) {
    int i = blockIdx.x * blockDim.x + threadIdx.x;
    if (i >= n_nodes) return;
    float ut  = x_t[3 * (size_t)i];                               // last use
    float ut1 = __builtin_nontemporal_load(&x_t1[3 * (size_t)i]);
    v4f   r   = __builtin_nontemporal_load(&acc4[i]);             // sums,cnt,se
    float m   = __builtin_nontemporal_load(&mask[i]);
    node_combine(ut, ut1, r.x, r.y, r.z, m, out, i);
}

// Generic fallback (rec_stride == 3, tight workspace).
__global__ void node_loss_kernel(const float* __restrict__ x_t,
                                 const float* __restrict__ x_t1,
                                 const float* __restrict__ mask,
                                 const float* __restrict__ acc,
                                 int rec_stride,
                                 float* __restrict__ out, int n_nodes) {
    int i = blockIdx.x * blockDim.x + threadIdx.x;
    if (i >= n_nodes) return;
    float ut  = x_t[3 * (size_t)i];
    float ut1 = __builtin_nontemporal_load(&x_t1[3 * (size_t)i]);
    const float* rec = acc + (size_t)i * rec_stride;
    float sm = __builtin_nontemporal_load(rec + 0);
    float c  = __builtin_nontemporal_load(rec + 1);
    float se = __builtin_nontemporal_load(rec + 2);
    float m  = __builtin_nontemporal_load(&mask[i]);
    node_combine(ut, ut1, sm, c, se, m, out, i);
}

extern "C" void kernel_launch(void* const* d_in, const int* in_sizes, int n_in,
                              void* d_out, int out_size, void* d_ws, size_t ws_size,
                              hipStream_t stream) {
    // setup_inputs order: x_t (N,3) f32, x_t1 (N,3) f32, edge_attr (E,2) f32,
    //                     mask (N,1) f32, edge_index (2,E) int
    const float* x_t       = (const float*)d_in[0];
    const float* x_t1      = (const float*)d_in[1];
    const float* edge_attr = (const float*)d_in[2];
    const float* mask      = (const float*)d_in[3];
    const int*   edge_idx  = (const int*)d_in[4];

    const int n_nodes = in_sizes[0] / 3;
    const int n_edges = in_sizes[2] / 2;

    const int* src = edge_idx;            // edge_index[0, :]
    const int* dst = edge_idx + n_edges;  // edge_index[1, :]

    // Prefer 16B-aligned 4-float records; fall back to 3 if ws is tight.
    const size_t need4 = (size_t)4 * (size_t)n_nodes * sizeof(float);
    const int rec_stride = (ws_size >= need4) ? 4 : 3;

    float* acc = (float*)d_ws;
    float* out = (float*)d_out;

    const int B = 256;

    // Re-zero accumulators every call (harness does not re-poison d_ws).
    const int zn  = rec_stride * n_nodes;
    const int zn4 = zn >> 2;
    zero_ws4_kernel<<<(zn4 + B - 1) / B, B, 0, stream>>>(
        reinterpret_cast<v4f*>(acc), zn4);
    if (zn & 3) {  // deterministic per problem size (normally empty)
        zero_ws_tail_kernel<<<1, B, 0, stream>>>(acc, zn4 << 2, zn);
    }

    const int n_pairs = (n_edges + 1) / 2;
    edge_scatter_kernel<<<(n_pairs + B - 1) / B, B, 0, stream>>>(
        x_t, edge_attr, src, dst, acc, rec_stride, n_edges);

    if (rec_stride == 4) {
        node_loss4_kernel<<<(n_nodes + B - 1) / B, B, 0, stream>>>(
            x_t, x_t1, mask, reinterpret_cast<const v4f*>(acc), out, n_nodes);
    } else {
        node_loss_kernel<<<(n_nodes + B - 1) / B, B, 0, stream>>>(
            x_t, x_t1, mask, acc, rec_stride, out, n_nodes);
    }
}